// DiagonalLinear_80960133529733
// MI455X (gfx1250) — compile-verified
//
#include <hip/hip_runtime.h>
#include <hip/hip_bf16.h>

// DiagonalLinear: out[r, c] = x[r, c] * diagonal[c]
// 32768 x 4096 fp32. Pure HBM-streaming problem (~1.07 GB @ 23.3 TB/s => ~46 us floor).
//
// MI455X / gfx1250 (wave32) strategy:
//  - 1024-thread blocks (32 waves). Thread t owns channels [4t, 4t+3] of every row,
//    so its diagonal float4 lives in registers for the whole block.
//  - b128 NT loads/stores: wave32 covers a contiguous 512 B segment per access;
//    NT hints keep the 1 GB one-touch stream from churning the 192 MB L2.
//  - Row stride = 16 KB, so all 32 rows of a block are addressed via the signed
//    24-bit instruction offset from ONE base address (no address math in loop).
//  - Batches of 8 loads into distinct registers -> 8 b128 loads in flight per
//    thread (descending s_wait_loadcnt drain instead of per-load serialization).
//  - global_prefetch_b8 (GL2 priming) 8 rows ahead.

typedef __attribute__((ext_vector_type(4))) float v4f;

namespace {
constexpr int kChannels     = 4096;
constexpr int kRows         = 32768;
constexpr int kThreads      = 1024;                  // 1024 * 4 floats = one full row
constexpr int kRowsPerBlock = 32;
constexpr int kBlocks       = kRows / kRowsPerBlock; // 1024 blocks, no tail
constexpr int kBatch        = 8;                     // loads in flight per thread
constexpr int kPrefetchRows = 8;                     // 8 rows = 128 KB lookahead
}

__global__ __launch_bounds__(kThreads)
void diag_scale_kernel(const float* __restrict__ x,
                       const float* __restrict__ diag,
                       float* __restrict__ out) {
  const int c4 = threadIdx.x << 2;                   // channel base for this lane

  // Per-thread diagonal slice, loaded once; hot in L1/L2 after the first waves.
  const v4f d = *(const v4f*)(diag + c4);

  const size_t base = (size_t)blockIdx.x * kRowsPerBlock * kChannels + c4;
  const float* __restrict__ xp = x + base;
  float* __restrict__ op       = out + base;

#pragma unroll
  for (int rb = 0; rb < kRowsPerBlock; rb += kBatch) {
    v4f v[kBatch];

    // Issue a clause of 8 independent b128 NT loads (constant instruction offsets).
#pragma unroll
    for (int i = 0; i < kBatch; ++i) {
      const size_t off = (size_t)(rb + i) * kChannels;
      __builtin_prefetch(xp + off + (size_t)kPrefetchRows * kChannels, 0, 0);
      v[i] = __builtin_nontemporal_load((const v4f*)(xp + off));
    }

    // Drain: scale and stream out as each load lands.
#pragma unroll
    for (int i = 0; i < kBatch; ++i) {
      const size_t off = (size_t)(rb + i) * kChannels;
      __builtin_nontemporal_store(v[i] * d, (v4f*)(op + off));
    }
  }
}

extern "C" void kernel_launch(void* const* d_in, const int* in_sizes, int n_in,
                              void* d_out, int out_size, void* d_ws, size_t ws_size,
                              hipStream_t stream) {
  (void)in_sizes; (void)n_in; (void)out_size; (void)d_ws; (void)ws_size;

  const float* x    = (const float*)d_in[0];   // (32768, 4096) fp32
  const float* diag = (const float*)d_in[1];   // (4096,) fp32
  float* out        = (float*)d_out;           // (32768, 4096) fp32

  diag_scale_kernel<<<kBlocks, kThreads, 0, stream>>>(x, diag, out);
}